// AdaFusionBlock_86294482912098
// MI455X (gfx1250) — compile-verified
//
#include <hip/hip_runtime.h>
#include <math.h>

// MI455X / gfx1250: wave32, WMMA. We use V_WMMA_F32_16X16X4_F32 (exact fp32)
// since the block is L2/gather bound (~10 GFLOP total, ~40MB working set in
// 192MB L2) and fp32 WMMA keeps reference numerics.

typedef float v2f __attribute__((ext_vector_type(2)));
typedef float v8f __attribute__((ext_vector_type(8)));

#define HW   16384   // 128*128
#define WDIM 128

static __device__ __forceinline__ v8f wmma_f32(v2f a, v2f b, v8f c) {
  // args: (neg_a, A, neg_b, B, c_mod, C, reuse_a, reuse_b)
  return __builtin_amdgcn_wmma_f32_16x16x4_f32(false, a, false, b, (short)0, c,
                                               false, false);
}

// ---------------------------------------------------------------------------
// Kernel 1: 1x1 conv (x0 = w0 @ x + b0).  GEMM M=64, K=64, N=32768.
// One wave = 16 out-channels x 16 pixels.  8 waves / 256-thread block.
// ---------------------------------------------------------------------------
__global__ void k_conv1x1(const float* __restrict__ x, const float* __restrict__ w0,
                          const float* __restrict__ b0, float* __restrict__ x0out) {
  const int lane  = threadIdx.x & 31;
  const int gw    = (blockIdx.x * blockDim.x + threadIdx.x) >> 5;
  const int mtile = gw & 3;                 // 4 tiles of 16 out-channels
  const int ptile = gw >> 2;                // 2048 pixel tiles
  const int b     = ptile >> 10;
  const int p0    = (ptile & 1023) << 4;
  const int n     = lane & 15;
  const int ks    = (lane >> 4) << 1;       // k offset 0 or 2
  const int m     = mtile * 16 + (lane & 15);
  const float* xb = x + b * 64 * HW;

  v8f acc = {};
  for (int k = 0; k < 64; k += 4) {
    v2f a, bb;
    a.x  = w0[m * 64 + k + ks];
    a.y  = w0[m * 64 + k + ks + 1];
    bb.x = xb[(k + ks) * HW + p0 + n];
    bb.y = xb[(k + ks + 1) * HW + p0 + n];
    acc  = wmma_f32(a, bb, acc);
  }
  const int mo = (lane >> 4) * 8;
#pragma unroll
  for (int v = 0; v < 8; ++v) {
    int o = mtile * 16 + mo + v;
    x0out[(b * 64 + o) * HW + p0 + n] = acc[v] + b0[o];
  }
}

// ---------------------------------------------------------------------------
// Generic implicit-im2col 3x3 conv, pad=1.  Input channels may be split
// across two tensors (for the concat([x0,y]) case).
// EPI: 0 = bias only, 1 = bias+leakyReLU(0.2), 2 = bias + residual add.
// ---------------------------------------------------------------------------
template <int CIN, int COUT, int MTILES, int CSPLIT, int EPI>
__global__ void k_conv3x3(const float* __restrict__ inA, const float* __restrict__ inB,
                          const float* __restrict__ wt, const float* __restrict__ bias,
                          const float* __restrict__ res, float* __restrict__ outp) {
  const int lane  = threadIdx.x & 31;
  const int gw    = (blockIdx.x * blockDim.x + threadIdx.x) >> 5;
  const int mtile = gw % MTILES;
  const int ptile = gw / MTILES;
  const int b     = ptile >> 10;
  const int p0    = (ptile & 1023) << 4;
  const int n     = lane & 15;
  const int p     = p0 + n;
  const int h     = p >> 7;
  const int w     = p & 127;
  const int pr    = (lane >> 4) << 1;       // reduction pair offset 0/2
  const int mrow  = mtile * 16 + (lane & 15);

  v8f acc = {};
  for (int r0 = 0; r0 < CIN * 9; r0 += 4) {
    const int ra = r0 + pr;
    v2f a, bb;
    if ((COUT % 16 == 0) || (mrow < COUT)) {
      a.x = wt[mrow * (CIN * 9) + ra];
      a.y = wt[mrow * (CIN * 9) + ra + 1];
    } else {
      a.x = 0.f; a.y = 0.f;
    }
    float bv[2];
#pragma unroll
    for (int j = 0; j < 2; ++j) {
      const int rr = ra + j;
      const int c  = rr / 9;
      const int t  = rr - 9 * c;
      const int kh = t / 3;
      const int kw = t - 3 * kh;
      const int hh = h + kh - 1;
      const int ww = w + kw - 1;
      float v = 0.f;
      if (hh >= 0 && hh < 128 && ww >= 0 && ww < 128) {
        if (c < CSPLIT)
          v = inA[(b * CSPLIT + c) * HW + hh * WDIM + ww];
        else
          v = inB[(b * (CIN - CSPLIT) + (c - CSPLIT)) * HW + hh * WDIM + ww];
      }
      bv[j] = v;
    }
    bb.x = bv[0];
    bb.y = bv[1];
    acc  = wmma_f32(a, bb, acc);
  }

  const int mo = (lane >> 4) * 8;
#pragma unroll
  for (int v = 0; v < 8; ++v) {
    const int o = mtile * 16 + mo + v;
    if ((COUT % 16 == 0) || (o < COUT)) {
      float val = acc[v] + bias[o];
      if (EPI == 1) val = (val >= 0.f) ? val : 0.2f * val;
      if (EPI == 2) val += res[(b * COUT + o) * HW + p];
      outp[(b * COUT + o) * HW + p] = val;
    }
  }
}

// ---------------------------------------------------------------------------
// Deformable 3x3 conv.  One wave = 16 pixels x all 64 out channels
// (4 WMMA accumulators), so the bilinear setup per (tap,pixel) is computed
// once and reused across the whole channel reduction.
// Reduction order r = t*64 + c  =>  A[o][r] = w_dc[o*576 + c*9 + t].
// ---------------------------------------------------------------------------
__global__ void k_deform(const float* __restrict__ y, const float* __restrict__ om,
                         const float* __restrict__ x0, const float* __restrict__ wdc,
                         const float* __restrict__ bdc, float* __restrict__ outp) {
  const int lane    = threadIdx.x & 31;
  const int gw      = (blockIdx.x * blockDim.x + threadIdx.x) >> 5;  // 2048
  const int b       = gw >> 10;
  const int p0      = (gw & 1023) << 4;
  const int n       = lane & 15;
  const int p       = p0 + n;
  const int h       = p >> 7;
  const int w       = p & 127;
  const int pairSel = lane >> 4;
  const int mr      = lane & 15;

  const float* yb  = y + b * 64 * HW;
  const float* omb = om + b * 27 * HW;

  v8f acc0 = {}, acc1 = {}, acc2 = {}, acc3 = {};

  for (int t = 0; t < 9; ++t) {
    const int kh = t / 3;
    const int kw = t - 3 * kh;
    const float dy = omb[(2 * t) * HW + p];
    const float dx = omb[(2 * t + 1) * HW + p];
    const float mk = 2.0f / (1.0f + __expf(-omb[(18 + t) * HW + p]));
    const float ys = (float)(h - 1 + kh) + dy;
    const float xs = (float)(w - 1 + kw) + dx;
    const float y0f = floorf(ys), x0f = floorf(xs);
    const float ty = ys - y0f, tx = xs - x0f;
    const int y0i = (int)y0f, x0i = (int)x0f;
    const int y1i = y0i + 1, x1i = x0i + 1;
    const float vy0 = (y0i >= 0 && y0i < 128) ? 1.f : 0.f;
    const float vy1 = (y1i >= 0 && y1i < 128) ? 1.f : 0.f;
    const float vx0 = (x0i >= 0 && x0i < 128) ? 1.f : 0.f;
    const float vx1 = (x1i >= 0 && x1i < 128) ? 1.f : 0.f;
    const int y0c = min(max(y0i, 0), 127), y1c = min(max(y1i, 0), 127);
    const int x0c = min(max(x0i, 0), 127), x1c = min(max(x1i, 0), 127);
    const float w00 = (1.f - ty) * (1.f - tx) * vy0 * vx0 * mk;
    const float w01 = (1.f - ty) * tx * vy0 * vx1 * mk;
    const float w10 = ty * (1.f - tx) * vy1 * vx0 * mk;
    const float w11 = ty * tx * vy1 * vx1 * mk;
    const int i00 = y0c * WDIM + x0c, i01 = y0c * WDIM + x1c;
    const int i10 = y1c * WDIM + x0c, i11 = y1c * WDIM + x1c;

    for (int cb = 0; cb < 64; cb += 4) {
      const int c = cb + 2 * pairSel;
      const float* yc0 = yb + c * HW;
      const float* yc1 = yc0 + HW;
      v2f bb;
      bb.x = w00 * yc0[i00] + w01 * yc0[i01] + w10 * yc0[i10] + w11 * yc0[i11];
      bb.y = w00 * yc1[i00] + w01 * yc1[i01] + w10 * yc1[i10] + w11 * yc1[i11];
      const int rbase = c * 9 + t;   // a.x index; a.y = +9 (next channel)
      v2f a;
      a.x = wdc[(0  + mr) * 576 + rbase]; a.y = wdc[(0  + mr) * 576 + rbase + 9];
      acc0 = wmma_f32(a, bb, acc0);
      a.x = wdc[(16 + mr) * 576 + rbase]; a.y = wdc[(16 + mr) * 576 + rbase + 9];
      acc1 = wmma_f32(a, bb, acc1);
      a.x = wdc[(32 + mr) * 576 + rbase]; a.y = wdc[(32 + mr) * 576 + rbase + 9];
      acc2 = wmma_f32(a, bb, acc2);
      a.x = wdc[(48 + mr) * 576 + rbase]; a.y = wdc[(48 + mr) * 576 + rbase + 9];
      acc3 = wmma_f32(a, bb, acc3);
    }
  }

  const int mo = pairSel * 8;
#pragma unroll
  for (int v = 0; v < 8; ++v) {
    const int o = mo + v;
    outp[(b * 64 + o)      * HW + p] = acc0[v] + bdc[o]      + x0[(b * 64 + o)      * HW + p];
    outp[(b * 64 + 16 + o) * HW + p] = acc1[v] + bdc[16 + o] + x0[(b * 64 + 16 + o) * HW + p];
    outp[(b * 64 + 32 + o) * HW + p] = acc2[v] + bdc[32 + o] + x0[(b * 64 + 32 + o) * HW + p];
    outp[(b * 64 + 48 + o) * HW + p] = acc3[v] + bdc[48 + o] + x0[(b * 64 + 48 + o) * HW + p];
  }
}

// ---------------------------------------------------------------------------
extern "C" void kernel_launch(void* const* d_in, const int* in_sizes, int n_in,
                              void* d_out, int out_size, void* d_ws, size_t ws_size,
                              hipStream_t stream) {
  (void)in_sizes; (void)n_in; (void)out_size; (void)ws_size;
  const float* x    = (const float*)d_in[0];
  const float* y    = (const float*)d_in[1];
  const float* w0   = (const float*)d_in[2];
  const float* b0   = (const float*)d_in[3];
  const float* w_om = (const float*)d_in[4];
  const float* b_om = (const float*)d_in[5];
  const float* w_dc = (const float*)d_in[6];
  const float* b_dc = (const float*)d_in[7];
  const float* w1   = (const float*)d_in[8];
  const float* b1   = (const float*)d_in[9];
  const float* w2   = (const float*)d_in[10];
  const float* b2   = (const float*)d_in[11];
  float* out = (float*)d_out;

  // Workspace layout (bytes): x0 (8MB) | om (3.375MB) | out (8MB) | tmp1 (8MB)
  char* ws = (char*)d_ws;
  float* x0buf  = (float*)(ws);
  float* ombuf  = (float*)(ws + 8388608);
  float* outbuf = (float*)(ws + 8388608 + 3538944);
  float* tmp1   = (float*)(ws + 8388608 + 3538944 + 8388608);

  dim3 blk(256);  // 8 waves/block
  // 1) x0 = conv1x1(x):                4 mtiles * 2048 ptiles = 8192 waves
  k_conv1x1<<<1024, blk, 0, stream>>>(x, w0, b0, x0buf);
  // 2) om = conv3x3(concat(x0,y)):     2 mtiles (27->32) * 2048 = 4096 waves
  k_conv3x3<128, 27, 2, 64, 0><<<512, blk, 0, stream>>>(x0buf, y, w_om, b_om,
                                                        nullptr, ombuf);
  // 3) out = x0 + deform_conv(y, om):  2048 waves (each owns 64 oc x 16 px)
  k_deform<<<256, blk, 0, stream>>>(y, ombuf, x0buf, w_dc, b_dc, outbuf);
  // 4) tmp1 = leakyrelu(conv3x3(out, w1) + b1)
  k_conv3x3<64, 64, 4, 64, 1><<<1024, blk, 0, stream>>>(outbuf, nullptr, w1, b1,
                                                        nullptr, tmp1);
  // 5) final = conv3x3(tmp1, w2) + b2 + out
  k_conv3x3<64, 64, 4, 64, 2><<<1024, blk, 0, stream>>>(tmp1, nullptr, w2, b2,
                                                        outbuf, out);
}